// TorchMultiTokenAttention_24068996726784
// MI455X (gfx1250) — compile-verified
//
#include <hip/hip_runtime.h>

typedef float v2f __attribute__((ext_vector_type(2)));
typedef float v8f __attribute__((ext_vector_type(8)));

#define LDIM   2048
#define BATCH  2
#define CHN    16
#define NGRP   4
#define CPG    4          // channels per group
#define KW     11
#define PADW   5
#define TILE   16
#define NXT    2          // x-tiles per wave
#define WGX    (TILE*NXT) // 32 output columns per workgroup
#define NROWS  (BATCH*CHN*LDIM)   // 65536
#define PT_R   26         // 16 + 2*PAD rows of probs needed
#define PT_C   44         // 32 + 2*PAD = 42 cols, padded to 44 for K-blocks

// ---------------------------------------------------------------------------
// Kernel 1: per-row causal-masked softmax statistics (max, 1/sum(exp))
// One 256-thread block per row of the [B,C,L,L] score tensor.
// ---------------------------------------------------------------------------
__global__ __launch_bounds__(256) void row_stats_kernel(
    const float* __restrict__ scores,
    float* __restrict__ mArr,
    float* __restrict__ invArr)
{
    __shared__ float red[256];
    const int row = blockIdx.x;            // 0 .. NROWS-1
    const int i   = row & (LDIM - 1);      // causal row index within the LxL map
    const float* srow = scores + (size_t)row * LDIM;
    const int t = threadIdx.x;

    float v[8];
    float mx = -3.0e38f;
    #pragma unroll
    for (int k = 0; k < 8; ++k) {
        int j = t + k * 256;
        float x = srow[j];
        v[k] = x;
        if (j <= i && x > mx) mx = x;
    }
    red[t] = mx;
    __syncthreads();
    for (int s = 128; s > 0; s >>= 1) {
        if (t < s) red[t] = fmaxf(red[t], red[t + s]);
        __syncthreads();
    }
    const float m = red[0];
    __syncthreads();

    float sum = 0.0f;
    #pragma unroll
    for (int k = 0; k < 8; ++k) {
        int j = t + k * 256;
        sum += (j <= i) ? __expf(v[k] - m) : 0.0f;
    }
    red[t] = sum;
    __syncthreads();
    for (int s = 128; s > 0; s >>= 1) {
        if (t < s) red[t] += red[t + s];
        __syncthreads();
    }
    if (t == 0) {
        mArr[row]   = m;
        invArr[row] = 1.0f / red[0];
    }
}

// ---------------------------------------------------------------------------
// Kernel 2: fused softmax-prob reconstruction + grouped 11x11 conv + bias +
// causal re-mask via V_WMMA_F32_16X16X4_F32 (Toeplitz mapping along x).
// Block = 128 threads (4 waves). Each wave produces TWO 16x16 (y,x) output
// tiles for one output channel of the group; the Toeplitz B fragments are
// shared across the two x-tiles (built once per (ci,dy), 7 kb-blocks).
// Grid = (L/32, L/16, B*NGRP).
// ---------------------------------------------------------------------------
__global__ __launch_bounds__(128) void conv_wmma_kernel(
    const float* __restrict__ scores,
    const float* __restrict__ weight,   // [C][CPG][KW][KW]
    const float* __restrict__ bias,     // [C]
    const float* __restrict__ mArr,
    const float* __restrict__ invArr,
    float* __restrict__ out)
{
    const int xt = blockIdx.x, yt = blockIdx.y;
    const int bz = blockIdx.z;
    const int b  = bz >> 2;
    const int g  = bz & 3;
    const int xbase = xt * WGX;
    const int ybase = yt * TILE;

    const int tid   = threadIdx.x;
    const int lane  = tid & 31;
    const int wave  = tid >> 5;
    const int ocl   = wave;                 // output channel within group
    const int oc    = g * CPG + ocl;

    const int mrow  = lane & 15;            // M (A) / N (B,C,D) index for lane
    const int khalf = (lane >> 4) * 2;      // K base for this half-wave
    const int rowhi = (lane >> 4) << 3;     // C/D: +8 row offset for upper half

    float* outBase = out + (((size_t)(b * CHN + oc)) * LDIM + ybase) * LDIM + xbase;

    // Fully-masked (strictly upper-triangular) strip: write zeros, skip work.
    if (xbase > ybase + (TILE - 1)) {
        #pragma unroll
        for (int i = 0; i < 8; ++i) {
            size_t ro = (size_t)(i + rowhi) * LDIM;
            outBase[ro + mrow]        = 0.0f;
            outBase[ro + TILE + mrow] = 0.0f;
        }
        return;
    }

    __shared__ float pt[CPG][PT_R][PT_C];       // probability tile (halo incl.)
    __shared__ float wlds[CPG * CPG * KW * KW]; // this group's weights

    // Cooperative load of group weights: wlds[ol*484 + cil*121 + dy*11 + dx]
    for (int idx = tid; idx < CPG * CPG * KW * KW; idx += 128) {
        wlds[idx] = weight[(size_t)g * (CPG * CPG * KW * KW) + idx];
    }

    // Cooperative build of the probability tile from scores + row stats.
    for (int idx = tid; idx < CPG * PT_R * PT_C; idx += 128) {
        int ci  = idx / (PT_R * PT_C);
        int rem = idx - ci * (PT_R * PT_C);
        int r   = rem / PT_C;
        int c   = rem - r * PT_C;
        int gr  = ybase - PADW + r;
        int gc  = xbase - PADW + c;
        int ch  = g * CPG + ci;
        float val = 0.0f;
        if (gr >= 0 && gr < LDIM && gc >= 0 && gc <= gr && c < (WGX + 2 * PADW)) {
            size_t rowIdx = (size_t)(b * CHN + ch) * LDIM + gr;
            float sc = scores[rowIdx * LDIM + gc];
            val = __expf(sc - mArr[rowIdx]) * invArr[rowIdx];
        }
        pt[ci][r][c] = val;
    }
    __syncthreads();

    // D[y,x] += sum_j pt[ci][y+dy][j(+16)] * T[j,x],  T[j,x] = w[oc,ci,dy, j-x]
    v8f acc0 = {};
    v8f acc1 = {};
    for (int ci = 0; ci < CPG; ++ci) {
        for (int dy = 0; dy < KW; ++dy) {
            const int wbase = ocl * (CPG * KW * KW) + ci * (KW * KW) + dy * KW;

            // Toeplitz B fragments, shared by both x-tiles (N = mrow,
            // K rows = kb*4 + khalf + {0,1}).
            v2f bf[7];
            #pragma unroll
            for (int kb = 0; kb < 7; ++kb) {
                int dx0 = (kb * 4 + khalf) - mrow;
                int cl0 = dx0 < 0 ? 0 : (dx0 > 10 ? 10 : dx0);
                float w0 = wlds[wbase + cl0];
                bf[kb].x = (dx0 >= 0 && dx0 <= 10) ? w0 : 0.0f;
                int dx1 = dx0 + 1;
                int cl1 = dx1 < 0 ? 0 : (dx1 > 10 ? 10 : dx1);
                float w1 = wlds[wbase + cl1];
                bf[kb].y = (dx1 >= 0 && dx1 <= 10) ? w1 : 0.0f;
            }

            const float* arow = &pt[ci][mrow + dy][0];
            #pragma unroll
            for (int kb = 0; kb < 7; ++kb) {
                const int c0 = kb * 4 + khalf;
                v2f a0;                      // contiguous pair -> ds_load_b64
                a0.x = arow[c0];
                a0.y = arow[c0 + 1];
                acc0 = __builtin_amdgcn_wmma_f32_16x16x4_f32(
                    false, a0, false, bf[kb], (short)0, acc0, false, false);

                v2f a1;
                a1.x = arow[TILE + c0];
                a1.y = arow[TILE + c0 + 1];
                acc1 = __builtin_amdgcn_wmma_f32_16x16x4_f32(
                    false, a1, false, bf[kb], (short)0, acc1, false, false);
            }
        }
    }

    // Bias + causal re-mask + store (C/D layout: vgpr i -> rows i and i+8)
    const float bv = bias[oc];
    #pragma unroll
    for (int i = 0; i < 8; ++i) {
        int yy = i + rowhi;
        int gy = ybase + yy;
        int gx0 = xbase + mrow;
        int gx1 = gx0 + TILE;
        size_t ro = (size_t)yy * LDIM;
        outBase[ro + mrow]        = (gx0 <= gy) ? (acc0[i] + bv) : 0.0f;
        outBase[ro + TILE + mrow] = (gx1 <= gy) ? (acc1[i] + bv) : 0.0f;
    }
}

extern "C" void kernel_launch(void* const* d_in, const int* in_sizes, int n_in,
                              void* d_out, int out_size, void* d_ws, size_t ws_size,
                              hipStream_t stream) {
    const float* scores = (const float*)d_in[0];
    const float* weight = (const float*)d_in[1];
    const float* bias   = (const float*)d_in[2];
    float* out    = (float*)d_out;
    float* mArr   = (float*)d_ws;          // NROWS floats
    float* invArr = mArr + NROWS;          // NROWS floats (total 512 KB of ws)

    row_stats_kernel<<<NROWS, 256, 0, stream>>>(scores, mArr, invArr);

    dim3 grid(LDIM / WGX, LDIM / TILE, BATCH * NGRP);
    conv_wmma_kernel<<<grid, 128, 0, stream>>>(scores, weight, bias,
                                               mArr, invArr, out);
}